// EdgeAssignmentLineGNN_1520418422913
// MI455X (gfx1250) — compile-verified
//
#include <hip/hip_runtime.h>
#include <hip/hip_bf16.h>

typedef float v2f __attribute__((ext_vector_type(2)));
typedef float v8f __attribute__((ext_vector_type(8)));

#define HIDC 128
#define TILE_ROWS 16
#define LDS_STRIDE 132  // 128 + 4 pad: row offsets hit distinct LDS banks

// ---------------------------------------------------------------------------
// Zero a float buffer (float4 grid-stride).
// ---------------------------------------------------------------------------
__global__ void zero_f32(float* __restrict__ p, long n4) {
  long i = (long)blockIdx.x * blockDim.x + threadIdx.x;
  long stride = (long)gridDim.x * blockDim.x;
  float4 z = make_float4(0.f, 0.f, 0.f, 0.f);
  for (; i < n4; i += stride) {
    ((float4*)p)[i] = z;
  }
}

// ---------------------------------------------------------------------------
// segment_sum: agg[dst[e]] += X[src[e]]   (one wave32 per edge; lane owns a
// float4 slice of the 128-float row -> coalesced 512B gather, f32 atomics).
// ---------------------------------------------------------------------------
__global__ void scatter_add_edges(const float* __restrict__ X,
                                  const int* __restrict__ src,
                                  const int* __restrict__ dst,
                                  float* __restrict__ agg, int nE) {
  long gid = (long)blockIdx.x * blockDim.x + threadIdx.x;
  int e = (int)(gid >> 5);
  int lane = (int)(gid & 31);
  if (e >= nE) return;
  int s = src[e];
  int d = dst[e];
  const float4 v = *(const float4*)(X + (size_t)s * HIDC + lane * 4);
  float* out = agg + (size_t)d * HIDC + lane * 4;
  atomicAdd(out + 0, v.x);
  atomicAdd(out + 1, v.y);
  atomicAdd(out + 2, v.z);
  atomicAdd(out + 3, v.w);
}

// ---------------------------------------------------------------------------
// Fused GraphConv GEMM: Y = relu(A@Wrel + X@Wroot + b), A,X:[N,128], W:[128,128]
// One block (8 waves) per 16-row tile; wave w computes cols [16w,16w+16).
// fp32 WMMA 16x16x4 — full precision; compute is negligible vs. scatter BW.
// ---------------------------------------------------------------------------
__global__ __launch_bounds__(256) void graphconv_gemm(
    const float* __restrict__ A, const float* __restrict__ X,
    const float* __restrict__ Wrel, const float* __restrict__ Wroot,
    const float* __restrict__ bias, float* __restrict__ Y) {
  __shared__ float sA[TILE_ROWS * LDS_STRIDE];
  __shared__ float sX[TILE_ROWS * LDS_STRIDE];
  const int tid = threadIdx.x;
  const int m0 = blockIdx.x * TILE_ROWS;

  // Cooperative stage of both 16x128 activation tiles (float4 per thread).
  for (int idx = tid; idx < TILE_ROWS * (HIDC / 4); idx += 256) {
    int r = idx >> 5;   // / (128/4)
    int c4 = idx & 31;
    float4 a = *(const float4*)(A + (size_t)(m0 + r) * HIDC + c4 * 4);
    float4 x = *(const float4*)(X + (size_t)(m0 + r) * HIDC + c4 * 4);
    *(float4*)&sA[r * LDS_STRIDE + c4 * 4] = a;
    *(float4*)&sX[r * LDS_STRIDE + c4 * 4] = x;
  }
  __syncthreads();

  const int wave = tid >> 5;   // 0..7 -> output col tile
  const int lane = tid & 31;
  const int half = lane >> 4;  // K-half selector for A/B fragments
  const int l16 = lane & 15;   // M (for A) / N (for B)
  const int n0 = wave * 16;

  v8f c = {};
  for (int k0 = 0; k0 < HIDC; k0 += 4) {
    const int ka = k0 + half * 2;
    // A fragment: lane l16 holds row l16, K = ka, ka+1 (ISA 32-bit A 16x4 layout)
    v2f a = *(const v2f*)&sA[l16 * LDS_STRIDE + ka];
    // B fragment: lane l16 holds col n0+l16, K = ka, ka+1
    v2f b;
    b.x = Wrel[(size_t)ka * HIDC + n0 + l16];
    b.y = Wrel[(size_t)(ka + 1) * HIDC + n0 + l16];
    c = __builtin_amdgcn_wmma_f32_16x16x4_f32(false, a, false, b, (short)0, c,
                                              false, false);
    v2f a2 = *(const v2f*)&sX[l16 * LDS_STRIDE + ka];
    v2f b2;
    b2.x = Wroot[(size_t)ka * HIDC + n0 + l16];
    b2.y = Wroot[(size_t)(ka + 1) * HIDC + n0 + l16];
    c = __builtin_amdgcn_wmma_f32_16x16x4_f32(false, a2, false, b2, (short)0, c,
                                              false, false);
  }

  // C/D layout: VGPR r -> row (r + 8*half), col n0+l16.
  const int col = n0 + l16;
  const float bb = bias[col];
#pragma unroll
  for (int r = 0; r < 8; ++r) {
    int row = m0 + r + half * 8;
    float v = c[r] + bb;
    v = v > 0.f ? v : 0.f;
    Y[(size_t)row * HIDC + col] = v;
  }
}

// ---------------------------------------------------------------------------
// Dense GEMM: Y = act(X@W + b), X:[N,128], W:[128,Hout], Hout in {128,32}.
// blockDim = 32 * (Hout/16); one block per 16-row tile.
// ---------------------------------------------------------------------------
__global__ void dense_gemm(const float* __restrict__ X,
                           const float* __restrict__ W,
                           const float* __restrict__ bias,
                           float* __restrict__ Y, int Hout, int doRelu) {
  __shared__ float sX[TILE_ROWS * LDS_STRIDE];
  const int tid = threadIdx.x;
  const int m0 = blockIdx.x * TILE_ROWS;

  for (int idx = tid; idx < TILE_ROWS * (HIDC / 4); idx += blockDim.x) {
    int r = idx >> 5;
    int c4 = idx & 31;
    float4 x = *(const float4*)(X + (size_t)(m0 + r) * HIDC + c4 * 4);
    *(float4*)&sX[r * LDS_STRIDE + c4 * 4] = x;
  }
  __syncthreads();

  const int wave = tid >> 5;
  const int lane = tid & 31;
  const int half = lane >> 4;
  const int l16 = lane & 15;
  const int n0 = wave * 16;

  v8f c = {};
  for (int k0 = 0; k0 < HIDC; k0 += 4) {
    const int ka = k0 + half * 2;
    v2f a = *(const v2f*)&sX[l16 * LDS_STRIDE + ka];
    v2f b;
    b.x = W[(size_t)ka * Hout + n0 + l16];
    b.y = W[(size_t)(ka + 1) * Hout + n0 + l16];
    c = __builtin_amdgcn_wmma_f32_16x16x4_f32(false, a, false, b, (short)0, c,
                                              false, false);
  }

  const int col = n0 + l16;
  const float bb = bias[col];
#pragma unroll
  for (int r = 0; r < 8; ++r) {
    int row = m0 + r + half * 8;
    float v = c[r] + bb;
    if (doRelu) v = v > 0.f ? v : 0.f;
    Y[(size_t)row * Hout + col] = v;
  }
}

// ---------------------------------------------------------------------------
// Host-side orchestration.
// ---------------------------------------------------------------------------
extern "C" void kernel_launch(void* const* d_in, const int* in_sizes, int n_in,
                              void* d_out, int out_size, void* d_ws,
                              size_t ws_size, hipStream_t stream) {
  const float* x = (const float*)d_in[0];
  const int* edge_index = (const int*)d_in[1];
  const float* Wrel0 = (const float*)d_in[2];
  const float* brel0 = (const float*)d_in[3];
  const float* Wroot0 = (const float*)d_in[4];
  const float* Wrel1 = (const float*)d_in[5];
  const float* brel1 = (const float*)d_in[6];
  const float* Wroot1 = (const float*)d_in[7];
  const float* Wrel2 = (const float*)d_in[8];
  const float* brel2 = (const float*)d_in[9];
  const float* Wroot2 = (const float*)d_in[10];
  const float* Wc0 = (const float*)d_in[11];
  const float* bc0 = (const float*)d_in[12];
  const float* Wc1 = (const float*)d_in[13];
  const float* bc1 = (const float*)d_in[14];
  float* logits = (float*)d_out;

  const int nNodes = in_sizes[0] / HIDC;            // 100000
  const int nE = in_sizes[1] / 2;                   // 1600000
  const int* src = edge_index;
  const int* dst = edge_index + nE;

  const size_t featElems = (size_t)nNodes * HIDC;   // 12.8M floats
  float* agg = (float*)d_ws;
  float* bufA = agg + featElems;
  float* bufB = bufA + featElems;

  const int rowTiles = nNodes / TILE_ROWS;          // 6250 (exact)
  const long zeroN4 = (long)(featElems / 4);
  const int zeroBlocks = 2048;
  const int scatterBlocks = (int)(((long)nE * 32 + 255) / 256);

  // ---- layer 0: in -> hid ----
  zero_f32<<<zeroBlocks, 256, 0, stream>>>(agg, zeroN4);
  scatter_add_edges<<<scatterBlocks, 256, 0, stream>>>(x, src, dst, agg, nE);
  graphconv_gemm<<<rowTiles, 256, 0, stream>>>(agg, x, Wrel0, Wroot0, brel0,
                                               bufA);
  // ---- layer 1 ----
  zero_f32<<<zeroBlocks, 256, 0, stream>>>(agg, zeroN4);
  scatter_add_edges<<<scatterBlocks, 256, 0, stream>>>(bufA, src, dst, agg, nE);
  graphconv_gemm<<<rowTiles, 256, 0, stream>>>(agg, bufA, Wrel1, Wroot1, brel1,
                                               bufB);
  // ---- layer 2 ----
  zero_f32<<<zeroBlocks, 256, 0, stream>>>(agg, zeroN4);
  scatter_add_edges<<<scatterBlocks, 256, 0, stream>>>(bufB, src, dst, agg, nE);
  graphconv_gemm<<<rowTiles, 256, 0, stream>>>(agg, bufB, Wrel2, Wroot2, brel2,
                                               bufA);
  // ---- classifier head ----
  dense_gemm<<<rowTiles, 256, 0, stream>>>(bufA, Wc0, bc0, bufB, HIDC, 1);
  dense_gemm<<<rowTiles, 64, 0, stream>>>(bufB, Wc1, bc1, logits, 32, 0);
}